// ClassCapsule_42949672961644
// MI455X (gfx1250) — compile-verified
//
#include <hip/hip_runtime.h>
#include <hip/hip_bf16.h>

// Problem constants (from reference)
#define B_      32
#define G_      32      // IN_CH
#define DIM_    16      // IN_DIM
#define K_      10      // CLASSES
#define OD_     16      // OUT_DIM
#define H_      14
#define W_      14
#define N_      196     // H*W
#define OKD_    160     // CLASSES*OUT_DIM
#define NCAP_   6272    // G_*N_
#define NPAD_   208     // 13 tiles of 16
#define COORD_SCALE 0.1f
#define TWO_PI  6.2832f // 2*3.1416 per reference

typedef float v2f __attribute__((ext_vector_type(2)));
typedef float v8f __attribute__((ext_vector_type(8)));

// ---------------------------------------------------------------------------
// Kernel 1: votes[b][g][n][o] = Wc[g](160x16) @ vec[b][g](16x196) + bc[g]
// One block per (g,b); 4 waves; WMMA f32 16x16x4 chained over K=16.
// ---------------------------------------------------------------------------
__global__ __launch_bounds__(128) void votes_kernel(
    const float* __restrict__ x,     // (B, G*17, 14, 14)
    const float* __restrict__ Wc,    // (G, 160, 16)
    const float* __restrict__ bc,    // (G, 160)
    float* __restrict__ votes)       // (B, G, 196, 160)
{
    const int g = blockIdx.x;
    const int b = blockIdx.y;

    __shared__ float As[OKD_][DIM_];   // A: 160x16
    __shared__ float Bs[DIM_][NPAD_];  // B: 16x208 (zero-padded cols)
    __shared__ float bcs[OKD_];

    const int tid = threadIdx.x;

    const float* Wg = Wc + (size_t)g * OKD_ * DIM_;
    for (int idx = tid; idx < OKD_ * DIM_; idx += 128)
        As[idx >> 4][idx & 15] = Wg[idx];
    for (int idx = tid; idx < OKD_; idx += 128)
        bcs[idx] = bc[(size_t)g * OKD_ + idx];

    // vec[d][n] = x[b][g*17 + 1 + d][h][w] (+ coordinate offsets on d=0,1)
    const float* xb = x + ((size_t)b * (G_ * 17) + (size_t)g * 17) * N_;
    for (int idx = tid; idx < DIM_ * N_; idx += 128) {
        int d = idx / N_, n = idx % N_;
        float v = xb[(size_t)(1 + d) * N_ + n];
        int hh = n / W_, ww = n % W_;
        if (d == 0) v += (float)hh * COORD_SCALE;
        if (d == 1) v += (float)ww * COORD_SCALE;
        Bs[d][n] = v;
    }
    for (int idx = tid; idx < DIM_ * (NPAD_ - N_); idx += 128)
        Bs[idx / (NPAD_ - N_)][N_ + idx % (NPAD_ - N_)] = 0.0f;
    __syncthreads();

    const int wave = tid >> 5;
    const int lane = tid & 31;
    const int half = lane >> 4;   // lanes 0-15 -> K {0,1}; 16-31 -> K {2,3}
    const int l16  = lane & 15;

    // 10 o-tiles x 13 n-tiles, round-robin over 4 waves
    for (int t = wave; t < 10 * 13; t += 4) {
        const int ot = t / 13, nt = t % 13;
        const int row = ot * 16 + l16;   // A-frag M per lane
        const int col = nt * 16 + l16;   // B-frag / D-frag N per lane

        v8f c = {};
#pragma unroll
        for (int kk = 0; kk < DIM_; kk += 4) {
            const int ka = kk + half * 2;
            v2f a, bb;
            a.x  = As[row][ka];
            a.y  = As[row][ka + 1];
            bb.x = Bs[ka][col];
            bb.y = Bs[ka + 1][col];
            // D = A(16x4) * B(4x16) + C, fp32 WMMA (CDNA5)
            c = __builtin_amdgcn_wmma_f32_16x16x4_f32(
                    false, a, false, bb, (short)0, c, false, false);
        }

        if (col < N_) {
            float* vout = votes + (((size_t)b * G_ + g) * N_ + col) * OKD_;
#pragma unroll
            for (int r = 0; r < 8; ++r) {
                const int o = ot * 16 + r + half * 8;  // D VGPR r <-> M=r / r+8
                vout[o] = c[r] + bcs[o];
            }
        }
    }
}

// ---------------------------------------------------------------------------
// Kernel 2: fused EM routing, one block (8 wave32s) per batch element.
// R lives in global scratch (per-block private region); stats in LDS.
// ---------------------------------------------------------------------------
__global__ __launch_bounds__(256) void routing_kernel(
    const float* __restrict__ x,
    const float* __restrict__ beta_v,
    const float* __restrict__ beta_a,
    const int*   __restrict__ lamda_p,
    const float* __restrict__ votes_all,  // (B, 6272, 160)
    float* __restrict__ R_all,            // (B, 6272, 10)
    float* __restrict__ out)              // (B, 10)
{
    const int b   = blockIdx.x;
    const int tid = threadIdx.x;
    const float lamda = (float)lamda_p[0];

    const float* votes = votes_all + (size_t)b * NCAP_ * OKD_;
    float*       R     = R_all     + (size_t)b * NCAP_ * K_;
    const float* xb    = x + (size_t)b * (G_ * 17) * N_;

    __shared__ float act_s[NCAP_];           // 25 KB
    __shared__ float red[8][17];             // per-wave partials
    __shared__ float sumr_s[K_];
    __shared__ float uh_s[K_][OD_];
    __shared__ float sig_s[K_][OD_];
    __shared__ float inv2sig_s[K_][OD_];
    __shared__ float coef_s[K_];
    __shared__ float ahat_s[K_];

    for (int i = tid; i < NCAP_; i += 256) {
        const int g = i / N_, n = i % N_;
        act_s[i] = xb[(size_t)(g * 17) * N_ + n];
    }
    for (int idx = tid; idx < NCAP_ * K_; idx += 256)
        R[idx] = 1.0f / (float)K_;
    __syncthreads();

    const int wave = tid >> 5;
    const int lane = tid & 31;

    for (int iter = 0; iter < 3; ++iter) {
        // ---- pass 1: sum_r[k], u_h[k][d] ----
        for (int k = 0; k < K_; ++k) {
            float acc[17];
#pragma unroll
            for (int v = 0; v < 17; ++v) acc[v] = 0.0f;
            for (int i = tid; i < NCAP_; i += 256) {
                const float rh = R[(size_t)i * K_ + k] * act_s[i];
                const float* vv = votes + (size_t)i * OKD_ + k * OD_;
                acc[16] += rh;
#pragma unroll
                for (int d = 0; d < OD_; ++d) acc[d] += rh * vv[d];
            }
#pragma unroll
            for (int v = 0; v < 17; ++v) {
                float s = acc[v];
                for (int off = 16; off > 0; off >>= 1)
                    s += __shfl_down(s, off, 32);
                if (lane == 0) red[wave][v] = s;
            }
            __syncthreads();
            if (tid < 17) {
                float s = 0.0f;
#pragma unroll
                for (int wv = 0; wv < 8; ++wv) s += red[wv][tid];
                if (tid == 16) sumr_s[k] = s;
                else           uh_s[k][tid] = s;   // numerator for now
            }
            __syncthreads();
        }
        if (tid < K_ * OD_) {
            const int k = tid >> 4, d = tid & 15;
            uh_s[k][d] /= sumr_s[k];
        }
        __syncthreads();

        // ---- pass 2: sigma[k][d] ----
        for (int k = 0; k < K_; ++k) {
            float acc[16];
#pragma unroll
            for (int v = 0; v < 16; ++v) acc[v] = 0.0f;
            for (int i = tid; i < NCAP_; i += 256) {
                const float rh = R[(size_t)i * K_ + k] * act_s[i];
                const float* vv = votes + (size_t)i * OKD_ + k * OD_;
#pragma unroll
                for (int d = 0; d < OD_; ++d) {
                    const float df = vv[d] - uh_s[k][d];
                    acc[d] += rh * df * df;
                }
            }
#pragma unroll
            for (int v = 0; v < 16; ++v) {
                float s = acc[v];
                for (int off = 16; off > 0; off >>= 1)
                    s += __shfl_down(s, off, 32);
                if (lane == 0) red[wave][v] = s;
            }
            __syncthreads();
            if (tid < 16) {
                float s = 0.0f;
#pragma unroll
                for (int wv = 0; wv < 8; ++wv) s += red[wv][tid];
                sig_s[k][tid] = s / sumr_s[k];
            }
            __syncthreads();
        }

        // ---- cost, a_hat, gaussian coefficients ----
        if (tid < K_) {
            const int k = tid;
            float slog = 0.0f, cost = 0.0f;
#pragma unroll
            for (int d = 0; d < OD_; ++d) {
                const float lg = __logf(sig_s[k][d]);
                slog += lg;
                cost += beta_v[k] + lg;
            }
            cost *= sumr_s[k];
            const float ah = 1.0f / (1.0f + __expf(-lamda * (beta_a[k] - cost)));
            ahat_s[k] = ah;
            // a_hat / sqrt((2pi)^16 * prod sigma)
            coef_s[k] = ah * __expf(-0.5f * (16.0f * __logf(TWO_PI) + slog));
        }
        if (tid < K_ * OD_) {
            const int k = tid >> 4, d = tid & 15;
            inv2sig_s[k][d] = 0.5f / sig_s[k][d];
        }
        __syncthreads();

        // ---- R update (skipped after last iteration) ----
        if (iter < 2) {
            for (int i = tid; i < NCAP_; i += 256) {
                const float* vv = votes + (size_t)i * OKD_;
                float num[K_];
                float tot = 0.0f;
#pragma unroll
                for (int k = 0; k < K_; ++k) {
                    float s = 0.0f;
#pragma unroll
                    for (int d = 0; d < OD_; ++d) {
                        const float df = vv[k * OD_ + d] - uh_s[k][d];
                        s += df * df * inv2sig_s[k][d];
                    }
                    const float nm = coef_s[k] * __expf(-s);
                    num[k] = nm;
                    tot += nm;
                }
                const float inv = 1.0f / tot;
#pragma unroll
                for (int k = 0; k < K_; ++k)
                    R[(size_t)i * K_ + k] = num[k] * inv;
            }
            __syncthreads();
        }
    }

    if (tid < K_) out[(size_t)b * K_ + tid] = ahat_s[tid];
}

// ---------------------------------------------------------------------------
extern "C" void kernel_launch(void* const* d_in, const int* in_sizes, int n_in,
                              void* d_out, int out_size, void* d_ws, size_t ws_size,
                              hipStream_t stream) {
    const float* x      = (const float*)d_in[0];
    const float* Wc     = (const float*)d_in[1];
    const float* bc     = (const float*)d_in[2];
    const float* beta_v = (const float*)d_in[3];
    const float* beta_a = (const float*)d_in[4];
    const int*   lamda  = (const int*)  d_in[5];

    const size_t votes_elems = (size_t)B_ * G_ * N_ * OKD_;   // 32.1M floats
    float* votes = (float*)d_ws;
    float* R     = (float*)d_ws + votes_elems;

    dim3 gv(G_, B_);
    votes_kernel<<<gv, 128, 0, stream>>>(x, Wc, bc, votes);
    routing_kernel<<<B_, 256, 0, stream>>>(x, beta_v, beta_a, lamda,
                                           votes, R, (float*)d_out);
}